// GCN_24300924961028
// MI455X (gfx1250) — compile-verified
//
#include <hip/hip_runtime.h>
#include <hip/hip_bf16.h>

// ---------------------------------------------------------------------------
// GCN forward on MI455X (gfx1250, wave32).
// GEMMs use V_WMMA_F32_16X16X4_F32 (full f32 precision matrix cores).
//   - ldb/ldc templated so all B loads / C stores use immediate offsets
//     (no v_mul_u64 address math in the inner loop).
//   - one wave computes a full 16 x Nn row strip: A fragment stays in
//     registers and is reused across all column tiles.
// Edge aggregation uses coalesced global f32 atomics; self-loops are fused
// into the bias/activation finalize pass (norm = dinv[i]^2).
// ---------------------------------------------------------------------------

typedef float v2f __attribute__((ext_vector_type(2)));
typedef float v8f __attribute__((ext_vector_type(8)));

#define TPB 256

// ---------------- fill ----------------
__global__ void fill_kernel(float* __restrict__ p, float v, int n) {
    int i = blockIdx.x * blockDim.x + threadIdx.x;
    if (i < n) p[i] = v;
}

// ---------------- degree accumulation: deg[dst[e]] += 1 ----------------
__global__ void deg_kernel(const int* __restrict__ dst, float* __restrict__ deg, int E) {
    int e = blockIdx.x * blockDim.x + threadIdx.x;
    if (e < E) atomicAdd(&deg[dst[e]], 1.0f);
}

// ---------------- dinv = rsqrt(deg), in place ----------------
__global__ void rsqrt_kernel(float* __restrict__ d, int n) {
    int i = blockIdx.x * blockDim.x + threadIdx.x;
    if (i < n) d[i] = rsqrtf(d[i]);
}

// ---------------- WMMA f32 GEMM: C[M,Nn] = A[M,K] @ B[K,Nn] (+bias) --------
// One wave per 16-row strip; sweeps all Nn/16 column tiles with the A
// fragment held in registers. K and Nn are compile-time so the whole
// K x col-tile loop nest fully unrolls into immediate-offset loads and
// back-to-back v_wmma_f32_16x16x4_f32.
// Requires M % 16 == 0 (true: 50000 and 512).
template <int K, int Nn>
__global__ void wmma_gemm_f32(const float* __restrict__ A,
                              const float* __restrict__ B,
                              float* __restrict__ C,
                              const float* __restrict__ bias,
                              int row_tiles) {
    constexpr int COLT = Nn / 16;
    int wave = (blockIdx.x * blockDim.x + threadIdx.x) >> 5;
    if (wave >= row_tiles) return;         // wave-uniform guard: EXEC stays all-1s
    int lane = threadIdx.x & 31;
    int half = lane >> 4;                  // 0: lanes 0-15, 1: lanes 16-31
    int lm   = lane & 15;
    int row0 = wave << 4;

    // A fragment base: element A[(row0+lm), k0 + 2*half + r] -> arow[k0 + r]
    const float* __restrict__ arow  = A + (size_t)(row0 + lm) * K + 2 * half;
    // B fragment base: element B[(k0 + 2*half + r), c*16 + lm] -> bbase[(k0+r)*Nn + c*16]
    const float* __restrict__ bbase = B + (size_t)(2 * half) * Nn + lm;

    v8f acc[COLT];
#pragma unroll
    for (int c = 0; c < COLT; ++c) acc[c] = (v8f){};

#pragma unroll
    for (int k0 = 0; k0 < K; k0 += 4) {
        v2f a;
        a.x = arow[k0];
        a.y = arow[k0 + 1];
#pragma unroll
        for (int c = 0; c < COLT; ++c) {
            v2f b;
            b.x = bbase[k0 * Nn + c * 16];
            b.y = bbase[(k0 + 1) * Nn + c * 16];
            acc[c] = __builtin_amdgcn_wmma_f32_16x16x4_f32(
                /*neg_a=*/false, a, /*neg_b=*/false, b,
                /*c_mod=*/(short)0, acc[c], /*reuse_a=*/false, /*reuse_b=*/false);
        }
    }

    // D layout: VGPR v -> M = row0 + v + 8*half, N = c*16 + lm
    float* __restrict__ crow = C + (size_t)(row0 + 8 * half) * Nn + lm;
#pragma unroll
    for (int c = 0; c < COLT; ++c) {
        float bv = bias ? bias[c * 16 + lm] : 0.0f;
#pragma unroll
        for (int v = 0; v < 8; ++v) {
            crow[(size_t)v * Nn + c * 16] = acc[c][v] + bv;
        }
    }
}

// ------------- edge scatter: out[dst[e],f] += h[src[e],f]*dinv[s]*dinv[d] ---
template <int F>
__global__ void scatter_kernel(const float* __restrict__ h,
                               const int* __restrict__ src,
                               const int* __restrict__ dst,
                               const float* __restrict__ dinv,
                               float* __restrict__ out, int E) {
    int gid = blockIdx.x * blockDim.x + threadIdx.x;
    if (gid >= E * F) return;
    int e = gid / F;
    int f = gid - e * F;
    int s = src[e];
    int d = dst[e];
    float val = h[(size_t)s * F + f] * (dinv[s] * dinv[d]);
    atomicAdd(&out[(size_t)d * F + f], val);
}

// ------------- finalize: agg += xw*dinv^2 + bias, optional ReLU -------------
template <int F, bool RELU>
__global__ void finalize_kernel(float* __restrict__ agg,
                                const float* __restrict__ xw,
                                const float* __restrict__ dinv,
                                const float* __restrict__ bias, int N) {
    int gid = blockIdx.x * blockDim.x + threadIdx.x;
    if (gid >= N * F) return;
    int i = gid / F;
    int f = gid - i * F;
    float di = dinv[i];
    float v = agg[gid] + xw[gid] * di * di + bias[f];
    agg[gid] = RELU ? fmaxf(v, 0.0f) : v;
}

// ------------- pooling ------------------------------------------------------
template <int F>
__global__ void pool_accum_kernel(const float* __restrict__ h,
                                  const int* __restrict__ batch,
                                  float* __restrict__ sums, int N) {
    int gid = blockIdx.x * blockDim.x + threadIdx.x;
    if (gid >= N * F) return;
    int i = gid / F;
    int f = gid - i * F;
    atomicAdd(&sums[(size_t)batch[i] * F + f], h[gid]);
}

__global__ void count_kernel(const int* __restrict__ batch, float* __restrict__ cnt, int N) {
    int i = blockIdx.x * blockDim.x + threadIdx.x;
    if (i < N) atomicAdd(&cnt[batch[i]], 1.0f);
}

template <int F>
__global__ void pool_div_kernel(float* __restrict__ sums, const float* __restrict__ cnt, int G) {
    int gid = blockIdx.x * blockDim.x + threadIdx.x;
    if (gid >= G * F) return;
    int g = gid / F;
    sums[gid] /= fmaxf(cnt[g], 1.0f);
}

static inline int cdiv(long long a, long long b) { return (int)((a + b - 1) / b); }

extern "C" void kernel_launch(void* const* d_in, const int* in_sizes, int n_in,
                              void* d_out, int out_size, void* d_ws, size_t ws_size,
                              hipStream_t stream) {
    (void)n_in; (void)ws_size;
    const float* x    = (const float*)d_in[0];
    const int*   edge = (const int*)d_in[1];
    const int*   batch= (const int*)d_in[2];
    const float* b1   = (const float*)d_in[4];
    const float* W1   = (const float*)d_in[3];
    const float* W2   = (const float*)d_in[5];
    const float* b2   = (const float*)d_in[6];
    const float* Wfc  = (const float*)d_in[7];
    const float* bfc  = (const float*)d_in[8];
    float* out = (float*)d_out;

    const int N   = in_sizes[2];         // 50000
    const int E   = in_sizes[1] / 2;     // 800000
    const int D   = 64;                  // input/hidden feature dim
    const int D2  = 128;                 // conv2 output dim
    const int HID = in_sizes[8];         // 64
    const int G   = out_size / HID;      // 512

    const int* src = edge;
    const int* dst = edge + E;

    // -------- workspace layout (floats) --------
    float* ws    = (float*)d_ws;
    float* xw1   = ws;                       // N*D
    float* agg1  = xw1  + (size_t)N * D;     // N*D   (becomes h1 after finalize)
    float* hw2   = agg1 + (size_t)N * D;     // N*D2
    float* agg2  = hw2  + (size_t)N * D2;    // N*D2  (becomes h2 after finalize)
    float* dinv  = agg2 + (size_t)N * D2;    // N     (deg -> dinv in place)
    float* sums  = dinv + (size_t)N;         // G*D2  (becomes pooled)
    float* cnt   = sums + (size_t)G * D2;    // G

    // -------- 1) init accumulators --------
    fill_kernel<<<cdiv((long long)N * D,  TPB), TPB, 0, stream>>>(agg1, 0.0f, N * D);
    fill_kernel<<<cdiv((long long)N * D2, TPB), TPB, 0, stream>>>(agg2, 0.0f, N * D2);
    fill_kernel<<<cdiv((long long)G * D2 + G, TPB), TPB, 0, stream>>>(sums, 0.0f, G * D2 + G);
    fill_kernel<<<cdiv(N, TPB), TPB, 0, stream>>>(dinv, 1.0f, N);   // self-loop degree

    // -------- 2) degrees + normalization --------
    deg_kernel<<<cdiv(E, TPB), TPB, 0, stream>>>(dst, dinv, E);
    rsqrt_kernel<<<cdiv(N, TPB), TPB, 0, stream>>>(dinv, N);

    // -------- 3) conv1: xW1 = x @ W1   (WMMA f32, K=64, Nn=64) --------
    {
        int row_tiles = N / 16;          // 3125 waves
        wmma_gemm_f32<64, 64><<<cdiv((long long)row_tiles * 32, TPB), TPB, 0, stream>>>(
            x, W1, xw1, nullptr, row_tiles);
    }
    scatter_kernel<64><<<cdiv((long long)E * D, TPB), TPB, 0, stream>>>(
        xw1, src, dst, dinv, agg1, E);
    finalize_kernel<64, true><<<cdiv((long long)N * D, TPB), TPB, 0, stream>>>(
        agg1, xw1, dinv, b1, N);   // agg1 = h1 = relu(agg + self + b1)

    // -------- 4) conv2: hW2 = h1 @ W2  (WMMA f32, K=64, Nn=128) --------
    {
        int row_tiles = N / 16;
        wmma_gemm_f32<64, 128><<<cdiv((long long)row_tiles * 32, TPB), TPB, 0, stream>>>(
            agg1, W2, hw2, nullptr, row_tiles);
    }
    scatter_kernel<128><<<cdiv((long long)E * D2, TPB), TPB, 0, stream>>>(
        hw2, src, dst, dinv, agg2, E);
    finalize_kernel<128, false><<<cdiv((long long)N * D2, TPB), TPB, 0, stream>>>(
        agg2, hw2, dinv, b2, N);   // agg2 = h2

    // -------- 5) global mean pool --------
    pool_accum_kernel<128><<<cdiv((long long)N * D2, TPB), TPB, 0, stream>>>(
        agg2, batch, sums, N);
    count_kernel<<<cdiv(N, TPB), TPB, 0, stream>>>(batch, cnt, N);
    pool_div_kernel<128><<<cdiv((long long)G * D2, TPB), TPB, 0, stream>>>(sums, cnt, G);

    // -------- 6) FC: out = pooled @ Wfc + bfc  (WMMA f32, K=128, Nn=64) -----
    {
        int row_tiles = G / 16;          // 32 waves
        wmma_gemm_f32<128, 64><<<cdiv((long long)row_tiles * 32, TPB), TPB, 0, stream>>>(
            sums, Wfc, out, bfc, row_tiles);
    }
}